// PhysicsLoss_32521492365738
// MI455X (gfx1250) — compile-verified
//
#include <hip/hip_runtime.h>
#include <hip/hip_bf16.h>

typedef __attribute__((ext_vector_type(2))) float v2f;
typedef __attribute__((ext_vector_type(8))) float v8f;

#define GRAVITY_C 9.81f
#define NBV 1024   // blocks for vertex pass (multiple of 64 for WMMA reduce)
#define NBE 2048   // blocks for element pass (multiple of 64)
#define TPB 256    // 8 waves per block on wave32

// ---------- wave32 / block reductions ----------
__device__ __forceinline__ float waveReduceSum(float v) {
    #pragma unroll
    for (int o = 16; o > 0; o >>= 1) v += __shfl_xor(v, o, 32);
    return v;
}

__device__ __forceinline__ float blockReduceSum(float v, float* smem /*>=8 floats*/) {
    const int lane = threadIdx.x & 31;
    const int wave = threadIdx.x >> 5;
    v = waveReduceSum(v);
    if (lane == 0) smem[wave] = v;
    __syncthreads();
    v = (threadIdx.x < (blockDim.x >> 5)) ? smem[threadIdx.x] : 0.0f;
    if (wave == 0) v = waveReduceSum(v);
    return v;   // valid in thread 0
}

// ---------- pass 1: vertex terms (inertia + gravity) ----------
__global__ __launch_bounds__(TPB) void vert_kernel(
        const float* __restrict__ pn, const float* __restrict__ pc,
        const float* __restrict__ pp, const float* __restrict__ mass,
        int V, float* __restrict__ iP, float* __restrict__ gP) {
    __shared__ float s1[8];
    __shared__ float s2[8];
    float li = 0.0f, lg = 0.0f;
    const int stride = gridDim.x * blockDim.x;
    for (int i = blockIdx.x * blockDim.x + threadIdx.x; i < V; i += stride) {
        const float m  = mass[i];
        const float ax = pn[3*i+0] + pp[3*i+0] - 2.0f * pc[3*i+0];
        const float ay = pn[3*i+1] + pp[3*i+1] - 2.0f * pc[3*i+1];
        const float az = pn[3*i+2] + pp[3*i+2] - 2.0f * pc[3*i+2];
        li = fmaf(m, fmaf(ax, ax, fmaf(ay, ay, az*az)), li);
        lg = fmaf(m, pc[3*i+1], lg);
    }
    const float bi = blockReduceSum(li, s1);
    const float bg = blockReduceSum(lg, s2);
    if (threadIdx.x == 0) { iP[blockIdx.x] = bi; gP[blockIdx.x] = bg; }
}

// ---------- pass 2: Neo-Hookean strain over tets ----------
__global__ __launch_bounds__(TPB) void elem_kernel(
        const float* __restrict__ pn, const int* __restrict__ elems,
        const float* __restrict__ vol, const float* __restrict__ Rinv,
        const float* __restrict__ lamp, const float* __restrict__ mup,
        int E, float* __restrict__ sP) {
    __shared__ float s1[8];
    const float lam = *lamp;
    const float mu  = *mup;
    float ls = 0.0f;
    const int stride = gridDim.x * blockDim.x;
    for (int e = blockIdx.x * blockDim.x + threadIdx.x; e < E; e += stride) {
        if (e + stride < E) {   // prefetch next grid-stride tile -> global_prefetch_b8
            __builtin_prefetch(&Rinv[9 * (e + stride)], 0, 0);
            __builtin_prefetch(&elems[4 * (e + stride)], 0, 0);
        }
        const int4 q = ((const int4*)elems)[e];          // B128 index load
        const float* p0 = pn + 3 * q.x;
        const float* p1 = pn + 3 * q.y;
        const float* p2 = pn + 3 * q.z;
        const float* p3 = pn + 3 * q.w;
        const float v0x = p0[0], v0y = p0[1], v0z = p0[2];
        // Ds columns: b = v1-v0, c = v2-v0, d = v3-v0
        const float bx = p1[0]-v0x, by = p1[1]-v0y, bz = p1[2]-v0z;
        const float cx = p2[0]-v0x, cy = p2[1]-v0y, cz = p2[2]-v0z;
        const float dx = p3[0]-v0x, dy = p3[1]-v0y, dz = p3[2]-v0z;
        float R[9];
        #pragma unroll
        for (int j = 0; j < 9; ++j) R[j] = Rinv[9 * e + j];
        // F[i][k] = b_i*R[0][k] + c_i*R[1][k] + d_i*R[2][k]
        float F00, F01, F02, F10, F11, F12, F20, F21, F22;
        F00 = fmaf(bx, R[0], fmaf(cx, R[3], dx*R[6]));
        F01 = fmaf(bx, R[1], fmaf(cx, R[4], dx*R[7]));
        F02 = fmaf(bx, R[2], fmaf(cx, R[5], dx*R[8]));
        F10 = fmaf(by, R[0], fmaf(cy, R[3], dy*R[6]));
        F11 = fmaf(by, R[1], fmaf(cy, R[4], dy*R[7]));
        F12 = fmaf(by, R[2], fmaf(cy, R[5], dy*R[8]));
        F20 = fmaf(bz, R[0], fmaf(cz, R[3], dz*R[6]));
        F21 = fmaf(bz, R[1], fmaf(cz, R[4], dz*R[7]));
        F22 = fmaf(bz, R[2], fmaf(cz, R[5], dz*R[8]));
        float det = F00 * (F11*F22 - F12*F21)
                  - F01 * (F10*F22 - F12*F20)
                  + F02 * (F10*F21 - F11*F20);
        det = fmaxf(det, 1e-8f);
        const float ld = __logf(det);
        const float tr = F00*F00 + F01*F01 + F02*F02
                       + F10*F10 + F11*F11 + F12*F12
                       + F20*F20 + F21*F21 + F22*F22;
        const float psi = 0.5f*lam*ld*ld - mu*ld + 0.5f*mu*(tr - 3.0f);
        ls = fmaf(psi, vol[e], ls);
    }
    const float bs = blockReduceSum(ls, s1);
    if (threadIdx.x == 0) sP[blockIdx.x] = bs;
}

// ---------- WMMA partial-sum reducer (single wave, EXEC all-ones) ----------
// A = ones(16x4), chained D = A*B + C => D[m][n] = sum_k B[k][n] (m-independent).
// Every partial lands in exactly one (k,n) slot, so summing VGPR0 of D over
// all 32 lanes gives 2x the grand total regardless of B's exact lane layout.
__device__ __forceinline__ float wmmaArraySum(const float* __restrict__ p, int chunks) {
#if __has_builtin(__builtin_amdgcn_wmma_f32_16x16x4_f32)
    v2f a; a[0] = 1.0f; a[1] = 1.0f;
    v8f c = {};
    const int lane = threadIdx.x;   // 0..31, uniform control flow
    for (int i = 0; i < chunks; ++i) {
        v2f b;
        b[0] = p[i * 64 + lane];
        b[1] = p[i * 64 + 32 + lane];
        c = __builtin_amdgcn_wmma_f32_16x16x4_f32(
                /*neg_a=*/false, a, /*neg_b=*/false, b,
                /*c_mod=*/(short)0, c, /*reuse_a=*/false, /*reuse_b=*/false);
    }
    float v = c[0];
    v = waveReduceSum(v);
    return 0.5f * v;
#else
    float v = 0.0f;
    for (int i = threadIdx.x; i < chunks * 64; i += 32) v += p[i];
    return waveReduceSum(v);
#endif
}

__global__ __launch_bounds__(32) void finalize_kernel(
        const float* __restrict__ iP, const float* __restrict__ gP,
        const float* __restrict__ sP, float* __restrict__ out, int V, int E) {
    const float si = wmmaArraySum(iP, NBV / 64);
    const float sg = wmmaArraySum(gP, NBV / 64);
    const float ss = wmmaArraySum(sP, NBE / 64);
    if (threadIdx.x == 0) {
        const float inertia = 0.5f * si / (3.0f * (float)V);  // mean over V*3
        const float gravity = -GRAVITY_C * sg / (float)V;     // mean over V
        const float strain  = ss / (float)E;                  // mean over E
        out[0] = 1.0f * inertia + 0.01f * gravity + 1.0f * strain;
    }
}

extern "C" void kernel_launch(void* const* d_in, const int* in_sizes, int n_in,
                              void* d_out, int out_size, void* d_ws, size_t ws_size,
                              hipStream_t stream) {
    const float* pn   = (const float*)d_in[0];  // pos_next (V,3)
    const float* pc   = (const float*)d_in[1];  // pos_curr (V,3)
    const float* pp   = (const float*)d_in[2];  // pos_prev (V,3)
    const float* mass = (const float*)d_in[3];  // (V,1)
    const int*   el   = (const int*)d_in[4];    // elements (E,4)
    const float* vol  = (const float*)d_in[5];  // (E,)
    const float* Rinv = (const float*)d_in[6];  // (E,3,3)
    const float* lamp = (const float*)d_in[7];  // scalar
    const float* mup  = (const float*)d_in[8];  // scalar
    const int V = in_sizes[3];
    const int E = in_sizes[5];

    float* ws = (float*)d_ws;
    float* iP = ws;               // NBV partials
    float* gP = ws + NBV;         // NBV partials
    float* sP = ws + 2 * NBV;     // NBE partials  (total 16 KB)

    vert_kernel<<<NBV, TPB, 0, stream>>>(pn, pc, pp, mass, V, iP, gP);
    elem_kernel<<<NBE, TPB, 0, stream>>>(pn, el, vol, Rinv, lamp, mup, E, sP);
    finalize_kernel<<<1, 32, 0, stream>>>(iP, gP, sP, (float*)d_out, V, E);
}